// ClipNormalProposal_21242908246680
// MI455X (gfx1250) — compile-verified
//
#include <hip/hip_runtime.h>

// out = clamp(x + sqrt(SIGMA)*noise, MIN_VAL, MAX_VAL)
// SIGMA = 1.0 -> sqrt folds to 1.0f at compile time.
// Pure HBM-streaming kernel: float4 (b128) non-temporal loads/stores,
// wave32-friendly 256-thread blocks.

typedef __attribute__((ext_vector_type(4))) float v4f;

#define SQRT_SIGMA 1.0f
#define MIN_VAL   -3.0f
#define MAX_VAL    3.0f

__global__ __launch_bounds__(256) void ClipNormalProposal_vec4_kernel(
    const v4f* __restrict__ x,
    const v4f* __restrict__ noise,
    v4f* __restrict__ out,
    long long n4)
{
    long long i = (long long)blockIdx.x * blockDim.x + threadIdx.x;
    const long long stride = (long long)gridDim.x * blockDim.x;
    for (; i < n4; i += stride) {
        // NT hint: data is touched exactly once (402 MB > 192 MB L2),
        // so bypass cache retention -> global_load_b128 with TH=NT.
        v4f xv = __builtin_nontemporal_load(&x[i]);
        v4f nv = __builtin_nontemporal_load(&noise[i]);
        v4f r;
#pragma unroll
        for (int k = 0; k < 4; ++k) {
            float s = __builtin_fmaf(nv[k], SQRT_SIGMA, xv[k]); // folds to add
            s = __builtin_fmaxf(s, MIN_VAL);   // v_max_num_f32
            s = __builtin_fminf(s, MAX_VAL);   // v_min_num_f32
            r[k] = s;
        }
        __builtin_nontemporal_store(r, &out[i]);
    }
}

// Scalar tail for n % 4 != 0 (not hit for 4096*8192, kept for generality).
__global__ __launch_bounds__(32) void ClipNormalProposal_tail_kernel(
    const float* __restrict__ x,
    const float* __restrict__ noise,
    float* __restrict__ out,
    long long start, long long n)
{
    long long i = start + threadIdx.x;
    if (i < n) {
        float s = __builtin_fmaf(noise[i], SQRT_SIGMA, x[i]);
        s = __builtin_fmaxf(s, MIN_VAL);
        s = __builtin_fminf(s, MAX_VAL);
        out[i] = s;
    }
}

extern "C" void kernel_launch(void* const* d_in, const int* in_sizes, int n_in,
                              void* d_out, int out_size, void* d_ws, size_t ws_size,
                              hipStream_t stream) {
    const float* x     = (const float*)d_in[0];
    const float* noise = (const float*)d_in[1];
    float* out         = (float*)d_out;

    const long long n  = (long long)in_sizes[0];   // 4096*8192 = 33,554,432
    const long long n4 = n >> 2;                   // 8,388,608 float4s

    if (n4 > 0) {
        const int block = 256;                     // 8 wave32s per block
        long long blocks = (n4 + block - 1) / block;
        // One float4 per thread; grid-stride guard keeps it correct if the
        // grid is ever clamped.
        ClipNormalProposal_vec4_kernel<<<dim3((unsigned int)blocks), dim3(block), 0, stream>>>(
            (const v4f*)x, (const v4f*)noise, (v4f*)out, n4);
    }

    const long long rem_start = n4 << 2;
    if (rem_start < n) {
        ClipNormalProposal_tail_kernel<<<dim3(1), dim3(32), 0, stream>>>(
            x, noise, out, rem_start, n);
    }
}